// pdhg_layer_19713899889095
// MI455X (gfx1250) — compile-verified
//
#include <hip/hip_runtime.h>

typedef float v2f __attribute__((ext_vector_type(2)));
typedef float v8f __attribute__((ext_vector_type(8)));

#define D 128

// ---------------------------------------------------------------------------
// Persistent dual/single-output fp32 WMMA GEMM:
//   O_j = Z @ W_j^T + B_j + coef_j * e_j[row],  coef_j = sgn_j * (*s_j)
// Block = 256 threads (8 wave32). Weights staged in LDS ONCE per block, then
// the block loops over 16-row tiles (tile += gridDim.x), restaging only the
// 8 KB Z tile. Wave w owns output columns [16w, 16w+16).
// LDS rows padded to 132 floats to avoid bank conflicts on stride-128 reads.
// DUAL is a template parameter so the K-loop has no runtime branches and the
// compiler can fully unroll + software-pipeline ds_load_b64 -> v_wmma.
// ---------------------------------------------------------------------------
template <bool DUAL>
__global__ __launch_bounds__(256) void gemm_wmma(
    const float* __restrict__ Z,
    const float* __restrict__ W1, const float* __restrict__ B1,
    float* __restrict__ O1, const float* __restrict__ e1,
    const float* __restrict__ s1, float sgn1,
    const float* __restrict__ W2, const float* __restrict__ B2,
    float* __restrict__ O2, const float* __restrict__ e2,
    const float* __restrict__ s2, float sgn2,
    int ntiles)
{
    __shared__ float Zs[16][132];
    __shared__ float W1s[128][132];
    __shared__ float W2s[DUAL ? 128 : 1][132];

    const int tid  = threadIdx.x;
    const int lane = tid & 31;
    const int wave = tid >> 5;

    // ---- Stage weights once per block: 128x128 = 4096 float4, 16/thread ----
    for (int i = tid; i < 4096; i += 256) {
        const int r  = i >> 5;
        const int c4 = i & 31;
        const float4 v = ((const float4*)(W1 + (size_t)r * D))[c4];
        W1s[r][c4 * 4 + 0] = v.x; W1s[r][c4 * 4 + 1] = v.y;
        W1s[r][c4 * 4 + 2] = v.z; W1s[r][c4 * 4 + 3] = v.w;
        if constexpr (DUAL) {
            const float4 u = ((const float4*)(W2 + (size_t)r * D))[c4];
            W2s[r][c4 * 4 + 0] = u.x; W2s[r][c4 * 4 + 1] = u.y;
            W2s[r][c4 * 4 + 2] = u.z; W2s[r][c4 * 4 + 3] = u.w;
        }
    }

    const int lh = lane >> 4;   // half-wave selects K pair (0 -> K+0/1, 1 -> K+2/3)
    const int ml = lane & 15;   // M (A frag) / N (B frag) index within tile
    const int n0 = wave * 16;
    const int col = n0 + ml;

    const float coef1 = s1 ? sgn1 * (*s1) : 0.0f;
    const float coef2 = (DUAL && s2) ? sgn2 * (*s2) : 0.0f;
    const float bias1 = B1[col];
    const float bias2 = DUAL ? B2[col] : 0.0f;

    for (int tile = blockIdx.x; tile < ntiles; tile += gridDim.x) {
        const int row0 = tile * 16;

        __syncthreads();  // weights staged / previous tile's Zs readers done

        // ---- Stage Z tile: 16x128 = 512 float4, 2 per thread ----
        for (int i = tid; i < 512; i += 256) {
            const int r  = i >> 5;
            const int c4 = i & 31;
            const float4 v = ((const float4*)(Z + (size_t)(row0 + r) * D))[c4];
            Zs[r][c4 * 4 + 0] = v.x; Zs[r][c4 * 4 + 1] = v.y;
            Zs[r][c4 * 4 + 2] = v.z; Zs[r][c4 * 4 + 3] = v.w;
        }
        __syncthreads();

        v8f c1 = {};
        v8f c2 = {};
        #pragma unroll
        for (int k = 0; k < D; k += 4) {
            const v2f a  = *(const v2f*)&Zs[ml][k + 2 * lh];
            const v2f b1 = *(const v2f*)&W1s[col][k + 2 * lh];
            c1 = __builtin_amdgcn_wmma_f32_16x16x4_f32(false, a, false, b1,
                                                       (short)0, c1, false, false);
            if constexpr (DUAL) {
                const v2f b2 = *(const v2f*)&W2s[col][k + 2 * lh];
                c2 = __builtin_amdgcn_wmma_f32_16x16x4_f32(false, a, false, b2,
                                                           (short)0, c2, false, false);
            }
        }

        #pragma unroll
        for (int r = 0; r < 8; ++r) {
            const int row = row0 + r + 8 * lh;  // C/D layout: lanes 0-15 M=r, 16-31 M=r+8
            const float ex1 = e1 ? coef1 * e1[row] : 0.0f;
            O1[(size_t)row * D + col] = c1[r] + bias1 + ex1;
            if constexpr (DUAL) {
                const float ex2 = e2 ? coef2 * e2[row] : 0.0f;
                O2[(size_t)row * D + col] = c2[r] + bias2 + ex2;
            }
        }
    }
}

// ---------------------------------------------------------------------------
// A^T SpMM accumulate:  Acc[cols[e]] += (*tau) * vals[e] * X[rows[e]]
// One edge per wave32; each lane owns a float4 slab of the 128-wide row.
// Destination rows are L2-resident -> global_atomic_add_f32 (no return).
// Fixed-distance prefetch of the sequential edge streams (HBM-streaming).
// ---------------------------------------------------------------------------
__global__ __launch_bounds__(256) void spmm_scatter_x(
    const int* __restrict__ rows, const int* __restrict__ cols,
    const float* __restrict__ vals, const float* __restrict__ X,
    float* __restrict__ Acc, const float* __restrict__ tau, int nnz)
{
    const int e = (int)((blockIdx.x * 256u + threadIdx.x) >> 5);
    if (e >= nnz) return;
    const int lane = threadIdx.x & 31;
    const int pf = e + 4096;
    if (pf < nnz) {
        __builtin_prefetch(&rows[pf], 0, 1);
        __builtin_prefetch(&cols[pf], 0, 1);
        __builtin_prefetch(&vals[pf], 0, 1);
    }
    const int src = rows[e];     // gather from X[rows[e]]
    const int dst = cols[e];     // scatter to Acc[cols[e]]
    const float sv = (*tau) * vals[e];
    const float4 xv = ((const float4*)(X + (size_t)src * D))[lane];
    float* o = Acc + (size_t)dst * D + lane * 4;
    atomicAdd(o + 0, sv * xv.x);
    atomicAdd(o + 1, sv * xv.y);
    atomicAdd(o + 2, sv * xv.z);
    atomicAdd(o + 3, sv * xv.w);
}

// ---------------------------------------------------------------------------
// Fused A SpMM pair:  Acc[rows[e]] += (*sigma) * vals[e] * (2*WX - VX)[cols[e]]
// Merges the a_wx and a_vx SpMMs into one gather/scatter pass over the edges.
// ---------------------------------------------------------------------------
__global__ __launch_bounds__(256) void spmm_scatter_y(
    const int* __restrict__ rows, const int* __restrict__ cols,
    const float* __restrict__ vals,
    const float* __restrict__ WX, const float* __restrict__ VX,
    float* __restrict__ Acc, const float* __restrict__ sigma, int nnz)
{
    const int e = (int)((blockIdx.x * 256u + threadIdx.x) >> 5);
    if (e >= nnz) return;
    const int lane = threadIdx.x & 31;
    const int pf = e + 4096;
    if (pf < nnz) {
        __builtin_prefetch(&rows[pf], 0, 1);
        __builtin_prefetch(&cols[pf], 0, 1);
        __builtin_prefetch(&vals[pf], 0, 1);
    }
    const int src = cols[e];     // gather
    const int dst = rows[e];     // scatter
    const float sv = (*sigma) * vals[e];
    const float4 w = ((const float4*)(WX + (size_t)src * D))[lane];
    const float4 v = ((const float4*)(VX + (size_t)src * D))[lane];
    float* o = Acc + (size_t)dst * D + lane * 4;
    atomicAdd(o + 0, sv * (2.0f * w.x - v.x));
    atomicAdd(o + 1, sv * (2.0f * w.y - v.y));
    atomicAdd(o + 2, sv * (2.0f * w.z - v.z));
    atomicAdd(o + 3, sv * (2.0f * w.w - v.w));
}

// ---------------------------------------------------------------------------
// out = relu(in), vectorized float4, grid-stride.
// ---------------------------------------------------------------------------
__global__ __launch_bounds__(256) void relu_copy(
    const float* __restrict__ in, float* __restrict__ out, int n4)
{
    for (int i = blockIdx.x * blockDim.x + threadIdx.x; i < n4;
         i += gridDim.x * blockDim.x) {
        float4 v = ((const float4*)in)[i];
        v.x = fmaxf(v.x, 0.0f);
        v.y = fmaxf(v.y, 0.0f);
        v.z = fmaxf(v.z, 0.0f);
        v.w = fmaxf(v.w, 0.0f);
        ((float4*)out)[i] = v;
    }
}

extern "C" void kernel_launch(void* const* d_in, const int* in_sizes, int n_in,
                              void* d_out, int out_size, void* d_ws, size_t ws_size,
                              hipStream_t stream)
{
    const float* x      = (const float*)d_in[0];
    const float* y      = (const float*)d_in[1];
    const float* c      = (const float*)d_in[2];
    const float* b      = (const float*)d_in[3];
    const float* A_vals = (const float*)d_in[4];
    const int*   A_rows = (const int*)d_in[5];
    const int*   A_cols = (const int*)d_in[6];
    const float* Ukx_w  = (const float*)d_in[7];
    const float* Ukx_b  = (const float*)d_in[8];
    const float* Uky_w  = (const float*)d_in[9];
    const float* Uky_b  = (const float*)d_in[10];
    const float* tau    = (const float*)d_in[11];
    const float* Vky_w  = (const float*)d_in[12];
    const float* Vky_b  = (const float*)d_in[13];
    const float* Wkx_w  = (const float*)d_in[14];
    const float* Wkx_b  = (const float*)d_in[15];
    const float* Vkx_w  = (const float*)d_in[16];
    const float* Vkx_b  = (const float*)d_in[17];
    const float* sigma  = (const float*)d_in[18];

    const int nnz   = in_sizes[4];
    const int nrows = in_sizes[2];          // N (c is [N,1])
    const size_t nd = (size_t)nrows * D;

    float* out_x = (float*)d_out;           // x_new
    float* out_y = out_x + nd;              // y_new

    float* buf0 = (float*)d_ws;             // uy, later wx
    float* buf1 = buf0 + nd;                // acc_x, later vx
    float* buf2 = buf1 + nd;                // acc_y

    const int ntiles = nrows / 16;          // 50000/16 = 3125 exactly
    const int gemm_blocks = ntiles < 800 ? ntiles : 800;  // persistent blocks
    const int sblocks = (nnz + 7) / 8;      // 8 edges (waves) per 256-thread block
    const int n4 = (int)(nd / 4);

    // 1) uy = Uky(y);  acc_y = Vky(y) - sigma*b     (reads y once)
    gemm_wmma<true><<<gemm_blocks, 256, 0, stream>>>(
        y, Uky_w, Uky_b, buf0, nullptr, nullptr, 0.0f,
        Vky_w, Vky_b, buf2, b, sigma, -1.0f, ntiles);

    // 2) acc_x = Ukx(x) - tau*c
    gemm_wmma<false><<<gemm_blocks, 256, 0, stream>>>(
        x, Ukx_w, Ukx_b, buf1, c, tau, -1.0f,
        nullptr, nullptr, nullptr, nullptr, nullptr, 0.0f, ntiles);

    // 3) acc_x += tau * (A^T @ uy)   (edge-parallel atomic scatter)
    spmm_scatter_x<<<sblocks, 256, 0, stream>>>(
        A_rows, A_cols, A_vals, buf0, buf1, tau, nnz);

    // 4) x_new = relu(acc_x)
    relu_copy<<<2048, 256, 0, stream>>>(buf1, out_x, n4);

    // 5) wx = Wkx(x_new);  vx = Vkx(x_new)          (reads x_new once)
    gemm_wmma<true><<<gemm_blocks, 256, 0, stream>>>(
        out_x, Wkx_w, Wkx_b, buf0, nullptr, nullptr, 0.0f,
        Vkx_w, Vkx_b, buf1, nullptr, nullptr, 0.0f, ntiles);

    // 6) acc_y += sigma * (A @ (2*wx - vx))         (single fused edge pass)
    spmm_scatter_y<<<sblocks, 256, 0, stream>>>(
        A_rows, A_cols, A_vals, buf0, buf1, buf2, sigma, nnz);

    // 7) y_new = relu(acc_y)
    relu_copy<<<2048, 256, 0, stream>>>(buf2, out_y, n4);
}